// GATLayer_70153995813485
// MI455X (gfx1250) — compile-verified
//
#include <hip/hip_runtime.h>

#define IN_DIM   64
#define E_DIM    32
#define K_MSG    96
#define K_APPLY  128
#define OUT_DIM  64
#define NEG_SLOPE 0.2f

typedef __attribute__((ext_vector_type(2))) float v2f;
typedef __attribute__((ext_vector_type(8))) float v8f;

// float atomic-max via int punning (memory initialized to -inf bit pattern)
__device__ __forceinline__ void atomic_max_f32(float* addr, float val) {
    if (val >= 0.0f) atomicMax((int*)addr, __float_as_int(val));
    else             atomicMin((unsigned int*)addr, __float_as_uint(val));
}

// ---------------------------------------------------------------- init ws
__global__ void k_init(float* __restrict__ z, float* __restrict__ nmax,
                       float* __restrict__ denom, int n_nodes) {
    long i = (long)blockIdx.x * blockDim.x + threadIdx.x;
    long tot = (long)n_nodes * OUT_DIM;
    if (i < tot) z[i] = 0.0f;
    if (i < n_nodes) { nmax[i] = -__builtin_inff(); denom[i] = 0.0f; }
}

// ------------------------------------------------- edge attention logits
// one wave32 per edge: coalesced float2 row loads + shfl reduction
__global__ void k_logits(const float* __restrict__ nfeats,
                         const int* __restrict__ src, const int* __restrict__ dst,
                         const float* __restrict__ attn_w, const float* __restrict__ attn_b,
                         float* __restrict__ elog, float* __restrict__ nmax,
                         int n_edges) {
    int  lane = threadIdx.x & 31;
    long e    = ((long)blockIdx.x * blockDim.x + threadIdx.x) >> 5;
    if (e >= n_edges) return;   // wave-uniform
    int s = src[e], d = dst[e];
    v2f hs = ((const v2f*)(nfeats + (long)s * IN_DIM))[lane];
    v2f hd = ((const v2f*)(nfeats + (long)d * IN_DIM))[lane];
    v2f a1 = ((const v2f*)attn_w)[lane];
    v2f a2 = ((const v2f*)(attn_w + IN_DIM))[lane];
    float v = hs.x * a1.x + hs.y * a1.y + hd.x * a2.x + hd.y * a2.y;
#pragma unroll
    for (int off = 16; off; off >>= 1) v += __shfl_xor(v, off, 32);
    v += attn_b[0];
    v = (v >= 0.0f) ? v : NEG_SLOPE * v;
    if (lane == 0) { elog[e] = v; atomic_max_f32(&nmax[d], v); }
}

// ------------------------------------------------ exp + segment denominator
__global__ void k_expsum(const int* __restrict__ dst, const float* __restrict__ nmax,
                         float* __restrict__ elog, float* __restrict__ denom, int n_edges) {
    long e = (long)blockIdx.x * blockDim.x + threadIdx.x;
    if (e >= n_edges) return;
    int d = dst[e];
    float ee = __expf(elog[e] - nmax[d]);
    elog[e] = ee;                    // overwrite logits with exp values
    atomicAdd(&denom[d], ee);
}

// --------------------- edge messages: [16 edges x 96] x [96 x 64] via f32 WMMA
// LDS weight layout: row kp = k/2 (128 dwords), k-pair interleaved + bank swizzle:
//   sW[kp*128 + ((2n + 32*(kp&1)) & 127) + (k&1)] = W[k][n]   (W[k][n] = proj_w[n*96+k])
// Read side: kp = 2*ks + half, so (kp&1)==half and column offsets are lane-constant;
// each B fragment is one aligned ds_load_b64 hitting all 64 banks conflict-free.
__global__ void __launch_bounds__(256)
k_message(const float* __restrict__ nfeats, const float* __restrict__ efeats,
          const int* __restrict__ src, const int* __restrict__ dst,
          const float* __restrict__ proj_w, const float* __restrict__ eexp,
          const float* __restrict__ denom, float* __restrict__ z,
          int n_edges, int n_tiles) {
    __shared__ float sW[K_MSG * OUT_DIM];    // 24 KB
    for (int i = threadIdx.x; i < K_MSG * OUT_DIM; i += blockDim.x) {
        int n = i / K_MSG, k = i - n * K_MSG;
        int kp = k >> 1;
        int c  = ((2 * n + ((kp & 1) << 5)) & 127) + (k & 1);
        sW[kp * 128 + c] = proj_w[i];
    }
    __syncthreads();

    int  lane = threadIdx.x & 31;
    int  half = lane >> 4, l16 = lane & 15;
    long tile = (long)blockIdx.x * (blockDim.x >> 5) + (threadIdx.x >> 5);
    if (tile >= n_tiles) return;     // wave-uniform

    long e0 = tile * 16;
    long er = e0 + l16; if (er >= n_edges) er = n_edges - 1;
    int   sv    = src[er];
    int   dv    = dst[er];
    float alpha = eexp[er] / denom[dv];
    if (e0 + l16 >= n_edges) alpha = 0.0f;   // dead tail rows contribute 0

    const float* arow = nfeats + (long)sv * IN_DIM;  // A rows = [h_src | efeat]
    const float* xrow = efeats + er * E_DIM;

    // lane-constant swizzled column offsets for the 4 N-tiles
    int cb = 2 * l16 + (half << 5);
    int c0 = cb & 127, c1 = (cb + 32) & 127, c2 = (cb + 64) & 127, c3 = (cb + 96) & 127;

    v8f acc0 = {0,0,0,0,0,0,0,0}, acc1 = {0,0,0,0,0,0,0,0};
    v8f acc2 = {0,0,0,0,0,0,0,0}, acc3 = {0,0,0,0,0,0,0,0};
#pragma unroll 4
    for (int ks = 0; ks < 24; ++ks) {        // K = 96, 4 per WMMA
        int k = ks * 4 + half * 2;           // A layout: lane-half carries K pairs
        v2f a;
        if (ks < 16) a = *(const v2f*)&arow[k];
        else         a = *(const v2f*)&xrow[k - 64];
        const float* srow = &sW[(ks * 2 + half) * 128];
        v2f b0 = *(const v2f*)&srow[c0];
        v2f b1 = *(const v2f*)&srow[c1];
        v2f b2 = *(const v2f*)&srow[c2];
        v2f b3 = *(const v2f*)&srow[c3];
        acc0 = __builtin_amdgcn_wmma_f32_16x16x4_f32(false, a, false, b0, (short)0, acc0, false, false);
        acc1 = __builtin_amdgcn_wmma_f32_16x16x4_f32(false, a, false, b1, (short)0, acc1, false, false);
        acc2 = __builtin_amdgcn_wmma_f32_16x16x4_f32(false, a, false, b2, (short)0, acc2, false, false);
        acc3 = __builtin_amdgcn_wmma_f32_16x16x4_f32(false, a, false, b3, (short)0, acc3, false, false);
    }

    // weighted scatter-add: D row m lives in VGPR r of lanes with (lane&15)=col, half = m>>3
#pragma unroll
    for (int r = 0; r < 8; ++r) {
        int   m  = r + half * 8;
        float am = __shfl(alpha, m, 32);
        int   dm = __shfl(dv,    m, 32);
        float* zr = z + (long)dm * OUT_DIM + l16;
        atomicAdd(zr,      am * acc0[r]);
        atomicAdd(zr + 16, am * acc1[r]);
        atomicAdd(zr + 32, am * acc2[r]);
        atomicAdd(zr + 48, am * acc3[r]);
    }
}

// --------------------- apply: relu([nfeats|z] @ apply_w^T + b) via f32 WMMA
__global__ void __launch_bounds__(256)
k_apply(const float* __restrict__ nfeats, const float* __restrict__ z,
        const float* __restrict__ apply_w, const float* __restrict__ apply_b,
        float* __restrict__ out, int n_nodes, int n_tiles) {
    __shared__ float sW[K_APPLY * OUT_DIM];   // 32 KB, same interleave+swizzle
    for (int i = threadIdx.x; i < K_APPLY * OUT_DIM; i += blockDim.x) {
        int o = i >> 7, k = i & 127;
        int kp = k >> 1;
        int c  = ((2 * o + ((kp & 1) << 5)) & 127) + (k & 1);
        sW[kp * 128 + c] = apply_w[i];
    }
    __syncthreads();

    int  lane = threadIdx.x & 31;
    int  half = lane >> 4, l16 = lane & 15;
    long tile = (long)blockIdx.x * (blockDim.x >> 5) + (threadIdx.x >> 5);
    if (tile >= n_tiles) return;

    long n0 = tile * 16;
    long nr = n0 + l16; if (nr >= n_nodes) nr = n_nodes - 1;
    const float* hrow = nfeats + nr * IN_DIM;
    const float* zrow = z      + nr * OUT_DIM;

    int cb = 2 * l16 + (half << 5);
    int c0 = cb & 127, c1 = (cb + 32) & 127, c2 = (cb + 64) & 127, c3 = (cb + 96) & 127;

    v8f acc0 = {0,0,0,0,0,0,0,0}, acc1 = {0,0,0,0,0,0,0,0};
    v8f acc2 = {0,0,0,0,0,0,0,0}, acc3 = {0,0,0,0,0,0,0,0};
#pragma unroll 4
    for (int ks = 0; ks < 32; ++ks) {         // K = 128
        int k = ks * 4 + half * 2;
        v2f a;
        if (ks < 16) a = *(const v2f*)&hrow[k];
        else         a = *(const v2f*)&zrow[k - 64];
        const float* srow = &sW[(ks * 2 + half) * 128];
        v2f b0 = *(const v2f*)&srow[c0];
        v2f b1 = *(const v2f*)&srow[c1];
        v2f b2 = *(const v2f*)&srow[c2];
        v2f b3 = *(const v2f*)&srow[c3];
        acc0 = __builtin_amdgcn_wmma_f32_16x16x4_f32(false, a, false, b0, (short)0, acc0, false, false);
        acc1 = __builtin_amdgcn_wmma_f32_16x16x4_f32(false, a, false, b1, (short)0, acc1, false, false);
        acc2 = __builtin_amdgcn_wmma_f32_16x16x4_f32(false, a, false, b2, (short)0, acc2, false, false);
        acc3 = __builtin_amdgcn_wmma_f32_16x16x4_f32(false, a, false, b3, (short)0, acc3, false, false);
    }

    float bb0 = apply_b[l16], bb1 = apply_b[l16 + 16];
    float bb2 = apply_b[l16 + 32], bb3 = apply_b[l16 + 48];
#pragma unroll
    for (int r = 0; r < 8; ++r) {
        long row = n0 + r + half * 8;
        if (row < n_nodes) {
            float* orow = out + row * OUT_DIM + l16;
            orow[0]  = fmaxf(acc0[r] + bb0, 0.0f);
            orow[16] = fmaxf(acc1[r] + bb1, 0.0f);
            orow[32] = fmaxf(acc2[r] + bb2, 0.0f);
            orow[48] = fmaxf(acc3[r] + bb3, 0.0f);
        }
    }
}

// ---------------------------------------------------------------- launcher
extern "C" void kernel_launch(void* const* d_in, const int* in_sizes, int n_in,
                              void* d_out, int out_size, void* d_ws, size_t ws_size,
                              hipStream_t stream) {
    const float* nfeats  = (const float*)d_in[0];
    const float* efeats  = (const float*)d_in[1];
    const int*   src     = (const int*)d_in[2];
    const int*   dst     = (const int*)d_in[3];
    const float* proj_w  = (const float*)d_in[4];
    const float* attn_w  = (const float*)d_in[5];
    const float* attn_b  = (const float*)d_in[6];
    const float* apply_w = (const float*)d_in[7];
    const float* apply_b = (const float*)d_in[8];
    float* out = (float*)d_out;

    int n_nodes = in_sizes[0] / IN_DIM;
    int n_edges = in_sizes[2];

    // workspace layout (floats): z[N*64] | nmax[N] | denom[N] | elog[E]   (~30.4 MB)
    float* z     = (float*)d_ws;
    float* nmax  = z + (long)n_nodes * OUT_DIM;
    float* denom = nmax + n_nodes;
    float* elog  = denom + n_nodes;

    long tot = (long)n_nodes * OUT_DIM;
    k_init<<<(int)((tot + 255) / 256), 256, 0, stream>>>(z, nmax, denom, n_nodes);

    long lthreads = (long)n_edges * 32;        // one wave32 per edge
    k_logits<<<(int)((lthreads + 255) / 256), 256, 0, stream>>>(
        nfeats, src, dst, attn_w, attn_b, elog, nmax, n_edges);

    k_expsum<<<(n_edges + 255) / 256, 256, 0, stream>>>(dst, nmax, elog, denom, n_edges);

    int etiles = (n_edges + 15) / 16;          // 8 waves (tiles) per block
    k_message<<<(etiles + 7) / 8, 256, 0, stream>>>(
        nfeats, efeats, src, dst, proj_w, elog, denom, z, n_edges, etiles);

    int ntiles = (n_nodes + 15) / 16;
    k_apply<<<(ntiles + 7) / 8, 256, 0, stream>>>(
        nfeats, z, apply_w, apply_b, out, n_nodes, ntiles);
}